// DFM_GAT_37641093382401
// MI455X (gfx1250) — compile-verified
//
#include <hip/hip_runtime.h>
#include <hip/hip_bf16.h>
#include <math.h>

// GAT forward on MI455X (gfx1250, wave32).
// Dense node-feature transform uses V_WMMA_F32_16X16X4_F32 (fp32-exact path),
// 4 M-tiles per wave to amortize B-fragment loads across 4 WMMAs per k-step.
// Edge softmax/aggregation is the memory-bound part: coalesced gathers + fp32
// atomics that stay resident in the 192MB L2.

#define NEG_SLOPE 0.2f
#define BN_INV 0.99999500003749968f   // 1/sqrt(1+1e-5)

typedef __attribute__((ext_vector_type(2))) float v2f;
typedef __attribute__((ext_vector_type(8))) float v8f;

// ---------- order-preserving float<->uint for atomic segment-max ----------
__device__ __forceinline__ unsigned ordEnc(float f) {
    unsigned u = __float_as_uint(f);
    return (u & 0x80000000u) ? ~u : (u | 0x80000000u);
}
__device__ __forceinline__ float ordDec(unsigned u) {
    return __uint_as_float((u & 0x80000000u) ? (u & 0x7FFFFFFFu) : ~u);
}

// ---------- fills ----------
__global__ __launch_bounds__(256) void k_fill_f32(float* p, int n, float v) {
    int i = blockIdx.x * 256 + threadIdx.x;
    if (i < n) p[i] = v;
}
__global__ __launch_bounds__(256) void k_fill_u32(unsigned* p, int n, unsigned v) {
    int i = blockIdx.x * 256 + threadIdx.x;
    if (i < n) p[i] = v;
}

// ---------- WMMA fp32 GEMM: H[N x 128] = X[N x K] @ W[K x 128] ----------
// Block = 256 threads = 8 waves, covers 64 rows x 128 cols.
// Wave w owns cols [16w,16w+16) and 4 row-tiles (t=0..3 -> rows 16t..16t+15).
// Per k-step: load one B fragment (2 cached global loads), reuse it for 4 WMMAs
// whose A fragments come from LDS (ds_load_b64, conflict-free stride 132).
// A frag layout (ISA 32-bit A 16x4): lanes 0-15 -> K{k0,k0+1}, lanes 16-31 -> K{k0+2,k0+3}.
template <int KPAD, bool KFULL>
__global__ __launch_bounds__(256) void k_gemm_wmma(const float* __restrict__ X,
                                                   const float* __restrict__ W,
                                                   float* __restrict__ H,
                                                   int N, int K) {
    constexpr int LDSW = KPAD + 4;            // 132 for K=128: 528B rows, 16B-aligned, bank-skewed
    __shared__ float As[64 * LDSW];
    const int rowBase = blockIdx.x * 64;

    if (KFULL) {                              // K == KPAD == 128: vectorized b128 staging
        for (int idx = threadIdx.x; idx < 64 * 32; idx += 256) {
            int r = idx >> 5, q = idx & 31;
            float4 v = make_float4(0.f, 0.f, 0.f, 0.f);
            int gr = rowBase + r;
            if (gr < N) v = ((const float4*)X)[(size_t)gr * 32 + q];
            *(float4*)&As[r * LDSW + q * 4] = v;
        }
    } else {                                  // small K: scalar staging, zero-padded to KPAD
        for (int idx = threadIdx.x; idx < 64 * KPAD; idx += 256) {
            int r = idx / KPAD, c = idx % KPAD;
            float v = 0.f;
            int gr = rowBase + r;
            if (gr < N && c < K) v = X[(size_t)gr * K + c];
            As[r * LDSW + c] = v;
        }
    }
    __syncthreads();

    const int wave  = threadIdx.x >> 5;
    const int lane  = threadIdx.x & 31;
    const int lhalf = lane >> 4;              // 0 or 1
    const int lm    = lane & 15;
    const int colBase = wave * 16;

    v8f acc[4];
#pragma unroll
    for (int t = 0; t < 4; ++t) acc[t] = (v8f){0.f, 0.f, 0.f, 0.f, 0.f, 0.f, 0.f, 0.f};

#pragma unroll 4
    for (int k0 = 0; k0 < KPAD; k0 += 4) {
        const int ka = k0 + lhalf * 2;
        v2f b;
        if (KFULL) {
            b.x = W[(size_t)ka * 128 + colBase + lm];
            b.y = W[(size_t)(ka + 1) * 128 + colBase + lm];
        } else {
            b.x = (ka < K)     ? W[(size_t)ka * 128 + colBase + lm]       : 0.f;
            b.y = (ka + 1 < K) ? W[(size_t)(ka + 1) * 128 + colBase + lm] : 0.f;
        }
#pragma unroll
        for (int t = 0; t < 4; ++t) {
            v2f a;
            a.x = As[(lm + t * 16) * LDSW + ka];
            a.y = As[(lm + t * 16) * LDSW + ka + 1];
            acc[t] = __builtin_amdgcn_wmma_f32_16x16x4_f32(
                /*neg_a=*/false, a, /*neg_b=*/false, b,
                /*c_mod=*/(short)0, acc[t], /*reuse_a=*/false, /*reuse_b=*/false);
        }
    }

    const int n = colBase + lm;
#pragma unroll
    for (int t = 0; t < 4; ++t) {
#pragma unroll
        for (int v = 0; v < 8; ++v) {         // lanes 0-15: M=v ; lanes 16-31: M=v+8
            int mrow = rowBase + t * 16 + v + lhalf * 8;
            if (mrow < N) H[(size_t)mrow * 128 + n] = acc[t][v];
        }
    }
}

// ---------- per-node attention logits: as/ad[n*2+h] = sum_c h[n,h,c]*a[h,c] ----------
__global__ __launch_bounds__(256) void k_alpha(const float* __restrict__ H,
                                               const float* __restrict__ a_src,
                                               const float* __restrict__ a_dst,
                                               float* __restrict__ as_out,
                                               float* __restrict__ ad_out, int N) {
    int i = blockIdx.x * 256 + threadIdx.x;     // i = n*2 + hh
    if (i >= N * 2) return;
    int n = i >> 1, hh = i & 1;
    const float* hr = H + (size_t)n * 128 + hh * 64;
    const float* av = a_src + hh * 64;
    const float* dv = a_dst + hh * 64;
    float ss = 0.f, dd = 0.f;
#pragma unroll 8
    for (int c = 0; c < 64; ++c) {
        float hv = hr[c];
        ss += hv * av[c];
        dd += hv * dv[c];
    }
    as_out[i] = ss;
    ad_out[i] = dd;
}

// ---------- edge pass A: e = leaky_relu(as[src]+ad[dst]); atomic segment max ----------
__global__ __launch_bounds__(256) void k_edge_logits(const int* __restrict__ ei, int E2, int N,
                                                     const float* __restrict__ as_,
                                                     const float* __restrict__ ad_,
                                                     float* __restrict__ p,
                                                     unsigned* __restrict__ m) {
    int i = blockIdx.x * 256 + threadIdx.x;     // i = e*2 + hh, e in [0, E2+N)
    int tot = (E2 + N) * 2;
    if (i >= tot) return;
    int e = i >> 1, hh = i & 1;
    int s, d;
    if (e < E2) { s = ei[e]; d = ei[E2 + e]; }
    else        { s = d = e - E2; }            // self-loop
    float v = as_[s * 2 + hh] + ad_[d * 2 + hh];
    v = (v > 0.f) ? v : NEG_SLOPE * v;
    p[i] = v;
    atomicMax(&m[d * 2 + hh], ordEnc(v));
}

// ---------- edge pass B: p = exp(e - m[dst]); atomic segment sum ----------
__global__ __launch_bounds__(256) void k_edge_exp(const int* __restrict__ ei, int E2, int N,
                                                  float* __restrict__ p,
                                                  const unsigned* __restrict__ m,
                                                  float* __restrict__ s) {
    int i = blockIdx.x * 256 + threadIdx.x;
    int tot = (E2 + N) * 2;
    if (i >= tot) return;
    int e = i >> 1, hh = i & 1;
    int d;
    if (e < E2) d = ei[E2 + e];
    else        d = e - E2;
    float mm = ordDec(m[d * 2 + hh]);
    float pe = expf(p[i] - mm);
    p[i] = pe;
    atomicAdd(&s[d * 2 + hh], pe);
}

// ---------- edge pass C: out[dst,c] += h[src,c] * alpha(edge, head(c)) ----------
// Consecutive threads share an edge and sweep c: gathers and atomics are
// fully coalesced 512B segments; destination accumulator is L2-resident.
__global__ __launch_bounds__(256) void k_edge_agg(const int* __restrict__ ei, int E2, int N,
                                                  const float* __restrict__ p,
                                                  const float* __restrict__ s,
                                                  const float* __restrict__ H,
                                                  float* __restrict__ out) {
    int i = blockIdx.x * 256 + threadIdx.x;     // i = e*128 + c, max 115.2M < 2^31
    int tot = (E2 + N) * 128;
    if (i >= tot) return;
    int e = i >> 7, c = i & 127, hh = c >> 6;
    int sn, d;
    if (e < E2) { sn = ei[e]; d = ei[E2 + e]; }
    else        { sn = d = e - E2; }
    float alpha = p[e * 2 + hh] / (s[d * 2 + hh] + 1e-16f);
    atomicAdd(&out[(size_t)d * 128 + c], H[(size_t)sn * 128 + c] * alpha);
}

// ---------- post: bias + BN(eval) + ELU ----------
__global__ __launch_bounds__(256) void k_post_concat(const float* __restrict__ agg,
                                                     const float* __restrict__ bias,
                                                     const float* __restrict__ gam,
                                                     const float* __restrict__ bet,
                                                     float* __restrict__ out, int N) {
    int i = blockIdx.x * 256 + threadIdx.x;
    if (i >= N * 128) return;
    int c = i & 127;
    float y = agg[i] + bias[c];
    y = y * (gam[c] * BN_INV) + bet[c];
    y = (y > 0.f) ? y : (expf(y) - 1.0f);
    out[i] = y;
}

__global__ __launch_bounds__(256) void k_post_final(const float* __restrict__ agg,
                                                    const float* __restrict__ bias,
                                                    const float* __restrict__ gam,
                                                    const float* __restrict__ bet,
                                                    float* __restrict__ out, int N) {
    int i = blockIdx.x * 256 + threadIdx.x;     // i = n*64 + c
    if (i >= N * 64) return;
    int n = i >> 6, c = i & 63;
    float y = 0.5f * (agg[(size_t)n * 128 + c] + agg[(size_t)n * 128 + 64 + c]) + bias[c];
    y = y * (gam[c] * BN_INV) + bet[c];
    y = (y > 0.f) ? y : (expf(y) - 1.0f);
    out[i] = y;
}

// ---------- pooling ----------
__global__ __launch_bounds__(256) void k_pool(const float* __restrict__ h,
                                              const int* __restrict__ batch,
                                              float* __restrict__ pool, int N) {
    int i = blockIdx.x * 256 + threadIdx.x;     // i = n*64 + c
    if (i >= N * 64) return;
    int n = i >> 6, c = i & 63;
    atomicAdd(&pool[batch[n] * 64 + c], h[i]);
}
__global__ __launch_bounds__(256) void k_count(const int* __restrict__ batch,
                                               float* __restrict__ cnt, int N) {
    int n = blockIdx.x * 256 + threadIdx.x;
    if (n >= N) return;
    atomicAdd(&cnt[batch[n]], 1.0f);
}

// ---------- MLP heads (B=64 graphs, one thread each) ----------
__global__ __launch_bounds__(64) void k_heads(const float* __restrict__ pool,
                                              const float* __restrict__ cnt,
                                              const float* __restrict__ gst,
                                              const float* __restrict__ rw1, const float* __restrict__ rb1,
                                              const float* __restrict__ rw2, const float* __restrict__ rb2,
                                              const float* __restrict__ cw1, const float* __restrict__ cb1,
                                              const float* __restrict__ cw2, const float* __restrict__ cb2,
                                              float* __restrict__ out, int B) {
    int b = blockIdx.x * blockDim.x + threadIdx.x;
    if (b >= B) return;
    float fused[75];
    float cc = fmaxf(cnt[b], 1.0f);
    for (int k = 0; k < 64; ++k) fused[k] = pool[b * 64 + k] / cc;
    for (int j = 0; j < 11; ++j) fused[64 + j] = gst[b * 11 + j];

    float yreg = rb2[0];
    for (int h = 0; h < 64; ++h) {
        float t = rb1[h];
        for (int k = 0; k < 75; ++k) t += fused[k] * rw1[k * 64 + h];
        t = fmaxf(t, 0.f);
        yreg += t * rw2[h];
    }
    out[b] = yreg;

    float yclf[5];
    for (int j = 0; j < 5; ++j) yclf[j] = cb2[j];
    for (int h = 0; h < 64; ++h) {
        float t = cb1[h];
        for (int k = 0; k < 75; ++k) t += fused[k] * cw1[k * 64 + h];
        t = fmaxf(t, 0.f);
        for (int j = 0; j < 5; ++j) yclf[j] += t * cw2[h * 5 + j];
    }
    for (int j = 0; j < 5; ++j) out[B + b * 5 + j] = yclf[j];
}

static inline int cdiv_i(long long a, long long b) { return (int)((a + b - 1) / b); }

extern "C" void kernel_launch(void* const* d_in, const int* in_sizes, int n_in,
                              void* d_out, int out_size, void* d_ws, size_t ws_size,
                              hipStream_t stream) {
    (void)n_in; (void)out_size; (void)ws_size;
    const float* x     = (const float*)d_in[0];
    const int*   ei    = (const int*)d_in[1];
    const int*   batch = (const int*)d_in[2];
    const float* gst   = (const float*)d_in[3];
    const int N  = in_sizes[0] / 9;
    const int E2 = in_sizes[1] / 2;
    const int B  = in_sizes[3] / 11;
    const int E  = E2 + N;

    const float *W[3], *Asrc[3], *Adst[3], *bias[3], *gam[3], *bet[3];
    for (int i = 0; i < 3; ++i) {
        W[i]    = (const float*)d_in[4 + i * 6 + 0];
        Asrc[i] = (const float*)d_in[4 + i * 6 + 1];
        Adst[i] = (const float*)d_in[4 + i * 6 + 2];
        bias[i] = (const float*)d_in[4 + i * 6 + 3];
        gam[i]  = (const float*)d_in[4 + i * 6 + 4];
        bet[i]  = (const float*)d_in[4 + i * 6 + 5];
    }
    const float* rw1 = (const float*)d_in[22];
    const float* rb1 = (const float*)d_in[23];
    const float* rw2 = (const float*)d_in[24];
    const float* rb2 = (const float*)d_in[25];
    const float* cw1 = (const float*)d_in[26];
    const float* cb1 = (const float*)d_in[27];
    const float* cw2 = (const float*)d_in[28];
    const float* cb2 = (const float*)d_in[29];

    // workspace layout
    float*    buf0 = (float*)d_ws;                     // N*128 : layer input/output
    float*    buf1 = buf0 + (size_t)N * 128;           // N*128 : h = X@W
    float*    buf2 = buf1 + (size_t)N * 128;           // N*128 : aggregation accumulator
    float*    asb  = buf2 + (size_t)N * 128;           // N*2
    float*    adb  = asb + (size_t)N * 2;              // N*2
    unsigned* mb   = (unsigned*)(adb + (size_t)N * 2); // N*2
    float*    sb   = (float*)mb + (size_t)N * 2;       // N*2
    float*    pb   = sb + (size_t)N * 2;               // E*2
    float*    pool = pb + (size_t)E * 2;               // B*64 (+B counts, contiguous)
    float*    cntb = pool + (size_t)B * 64;            // B

    const float* layer_in = x;
    for (int i = 0; i < 3; ++i) {
        if (i == 0)
            k_gemm_wmma<12, false><<<cdiv_i(N, 64), 256, 0, stream>>>(layer_in, W[i], buf1, N, 9);
        else
            k_gemm_wmma<128, true><<<cdiv_i(N, 64), 256, 0, stream>>>(layer_in, W[i], buf1, N, 128);
        k_alpha<<<cdiv_i((long long)N * 2, 256), 256, 0, stream>>>(buf1, Asrc[i], Adst[i], asb, adb, N);
        k_fill_u32<<<cdiv_i((long long)N * 2, 256), 256, 0, stream>>>(mb, N * 2, 0u);
        k_fill_f32<<<cdiv_i((long long)N * 2, 256), 256, 0, stream>>>(sb, N * 2, 0.f);
        k_fill_f32<<<cdiv_i((long long)N * 128, 256), 256, 0, stream>>>(buf2, N * 128, 0.f);
        k_edge_logits<<<cdiv_i((long long)E * 2, 256), 256, 0, stream>>>(ei, E2, N, asb, adb, pb, mb);
        k_edge_exp<<<cdiv_i((long long)E * 2, 256), 256, 0, stream>>>(ei, E2, N, pb, mb, sb);
        k_edge_agg<<<cdiv_i((long long)E * 128, 256), 256, 0, stream>>>(ei, E2, N, pb, sb, buf1, buf2);
        if (i < 2)
            k_post_concat<<<cdiv_i((long long)N * 128, 256), 256, 0, stream>>>(buf2, bias[i], gam[i], bet[i], buf0, N);
        else
            k_post_final<<<cdiv_i((long long)N * 64, 256), 256, 0, stream>>>(buf2, bias[i], gam[i], bet[i], buf0, N);
        layer_in = buf0;
    }

    k_fill_f32<<<cdiv_i((long long)B * 65, 256), 256, 0, stream>>>(pool, B * 65, 0.f);
    k_pool<<<cdiv_i((long long)N * 64, 256), 256, 0, stream>>>(buf0, batch, pool, N);
    k_count<<<cdiv_i(N, 256), 256, 0, stream>>>(batch, cntb, N);
    k_heads<<<1, 64, 0, stream>>>(pool, cntb, gst, rw1, rb1, rw2, rb2,
                                  cw1, cb1, cw2, cb2, (float*)d_out, B);
}